// Ternary_20083267076262
// MI455X (gfx1250) — compile-verified
//
#include <hip/hip_runtime.h>
#include <hip/hip_bf16.h>

// Census loss: x,y [8,3,256,256] f32 -> scalar f32.
// Tile: 32x32 outputs per block, 38x38 gray halo (pad=3) in LDS for each image.
// Hot loop uses raw v_rsq_f32 / v_rcp_f32 (loss math is approximation-tolerant).
// Wave-level reduction uses V_WMMA_F32_16X16X4_F32 with a ones B-matrix.

#define BATCH   8
#define HW      256
#define PLANE   (HW * HW)          // 65536
#define TILE    32
#define HALO    38                 // TILE + 2*3
#define LDSP    40                 // padded LDS row stride
#define NTHREAD 256

typedef float v2f __attribute__((ext_vector_type(2)));
typedef float v8f __attribute__((ext_vector_type(8)));

__global__ void census_zero_kernel(float* out) {
    out[0] = 0.0f;
}

__global__ __launch_bounds__(NTHREAD) void census_loss_kernel(
    const float* __restrict__ x, const float* __restrict__ y,
    float* __restrict__ out) {

    __shared__ float sgx[HALO * LDSP];
    __shared__ float sgy[HALO * LDSP];

    const int b   = blockIdx.z;
    const int ty0 = blockIdx.y * TILE;
    const int tx0 = blockIdx.x * TILE;
    const int gy0 = ty0 - 3;
    const int gx0 = tx0 - 3;

    const float* xb = x + (size_t)b * 3 * PLANE;
    const float* yb = y + (size_t)b * 3 * PLANE;

    // --- cooperative grayscale halo load (zero outside image == conv zero-pad)
    for (int i = threadIdx.x; i < HALO * HALO; i += NTHREAD) {
        const int ly = i / HALO;
        const int lx = i - ly * HALO;
        const int gy = gy0 + ly;
        const int gx = gx0 + lx;
        float vx = 0.0f, vy = 0.0f;
        if ((unsigned)gy < (unsigned)HW && (unsigned)gx < (unsigned)HW) {
            const int o = gy * HW + gx;
            vx = (xb[o] + xb[o + PLANE] + xb[o + 2 * PLANE]) * (1.0f / 3.0f);
            vy = (yb[o] + yb[o + PLANE] + yb[o + 2 * PLANE]) * (1.0f / 3.0f);
        }
        sgx[ly * LDSP + lx] = vx;
        sgy[ly * LDSP + lx] = vy;
    }
    __syncthreads();

    // --- per-thread: 4 output pixels, 49-tap census difference each
    const int lx  = threadIdx.x & (TILE - 1);
    const int ly0 = threadIdx.x >> 5;          // 0..7 (wave32: 8 rows of 32)

    // mean over 49 taps, mean over 8*1*256*256 masked map
    const float scale = 1.0f / (49.0f * 524288.0f);

    float acc = 0.0f;
    #pragma unroll
    for (int r = 0; r < 4; ++r) {
        const int ly = ly0 + r * 8;
        const int h  = ty0 + ly;
        const int w  = tx0 + lx;
        // valid-mask: inner region only; all its 7x7 windows are in-bounds
        if (h >= 3 && h <= HW - 4 && w >= 3 && w <= HW - 4) {
            const float cx = sgx[(ly + 3) * LDSP + (lx + 3)];
            const float cy = sgy[(ly + 3) * LDSP + (lx + 3)];
            #pragma unroll
            for (int dy = 0; dy < 7; ++dy) {
                #pragma unroll
                for (int dx = 0; dx < 7; ++dx) {
                    const float nx = sgx[(ly + dy) * LDSP + (lx + dx)];
                    const float ny = sgy[(ly + dy) * LDSP + (lx + dx)];
                    const float ax = nx - cx;
                    const float fx = ax * __builtin_amdgcn_rsqf(0.81f + ax * ax);
                    const float ay = ny - cy;
                    const float fy = ay * __builtin_amdgcn_rsqf(0.81f + ay * ay);
                    const float d  = fx - fy;
                    const float d2 = d * d;
                    acc += d2 * __builtin_amdgcn_rcpf(0.1f + d2);
                }
            }
        }
    }

    const float part = acc * scale;

    // --- wave32 reduction via V_WMMA_F32_16X16X4_F32 with B = ones.
    // D = A x Ones  =>  sum(D) = 16 * sum(A), independent of VGPR layout.
    v2f a; a[0] = part; a[1] = 0.0f;
    v2f bm; bm[0] = 1.0f; bm[1] = 1.0f;
    v8f c = {};
    c = __builtin_amdgcn_wmma_f32_16x16x4_f32(
        /*neg_a=*/false, a, /*neg_b=*/false, bm,
        /*c_mod=*/(short)0, c, /*reuse_a=*/false, /*reuse_b=*/false);

    float s = c[0] + c[1] + c[2] + c[3] + c[4] + c[5] + c[6] + c[7];
    #pragma unroll
    for (int off = 16; off >= 1; off >>= 1)
        s += __shfl_xor(s, off, 32);
    const float wave_total = s * 0.0625f;      // sum(D)/16 == sum(A)

    if ((threadIdx.x & 31) == 0)
        atomicAdd(out, wave_total);
}

extern "C" void kernel_launch(void* const* d_in, const int* in_sizes, int n_in,
                              void* d_out, int out_size, void* d_ws, size_t ws_size,
                              hipStream_t stream) {
    (void)in_sizes; (void)n_in; (void)out_size; (void)d_ws; (void)ws_size;
    const float* x = (const float*)d_in[0];
    const float* y = (const float*)d_in[1];
    float* out = (float*)d_out;

    census_zero_kernel<<<dim3(1), dim3(1), 0, stream>>>(out);

    dim3 grid(HW / TILE, HW / TILE, BATCH);   // 8 x 8 x 8
    dim3 block(NTHREAD);
    census_loss_kernel<<<grid, block, 0, stream>>>(x, y, out);
}